// GraphAttConv_48911087567195
// MI455X (gfx1250) — compile-verified
//
#include <hip/hip_runtime.h>
#include <hip/hip_bf16.h>
#include <math.h>

typedef __attribute__((ext_vector_type(2))) float v2f;
typedef __attribute__((ext_vector_type(8))) float v8f;

#define F_IN   256
#define F_OUT  256
#define HEADS  8
#define DHEAD  32
#define NEG_SLOPE 0.2f
#define GAT_EPS   1e-16f

// ---------- helpers ----------
__device__ __forceinline__ unsigned f_ord(float f) {
    unsigned u = __float_as_uint(f);
    return (u & 0x80000000u) ? ~u : (u | 0x80000000u);
}
__device__ __forceinline__ float f_unord(unsigned k) {
    return __uint_as_float((k & 0x80000000u) ? (k & 0x7fffffffu) : ~k);
}
__device__ __forceinline__ float lrelu(float z) { return z > 0.f ? z : NEG_SLOPE * z; }

// ---------- K1: H = x @ Wc  (Wc[k][h*32+d] = W[h][k][d]) ----------
// block = 256 threads = 8 waves; tile = 32 rows x 256 cols; K = 256
__global__ __launch_bounds__(256) void gat_gemm_wmma(
    const float* __restrict__ x, const float* __restrict__ W,
    float* __restrict__ H, int nrows)
{
    __shared__ float xs[32][260];  // padded stride 260 (260 % 64 == 4) -> conflict-free A reads
    const int tid = threadIdx.x;
    const int m0b = blockIdx.x * 32;

    // stage x tile (32 x 256 f32) straight into LDS via async-to-LDS b128 (ASYNCcnt path)
    {
        const unsigned ldsbase = (unsigned)(uintptr_t)&xs[0][0];  // low 32 bits of generic ptr = LDS offset
        for (int i = 0; i < 8; ++i) {
            int f   = i * 256 + tid;   // float4 index within tile [0, 2048)
            int row = f >> 6;          // 64 float4 per row
            int c4  = f & 63;
            int grow = m0b + row; if (grow > nrows - 1) grow = nrows - 1;
            unsigned lds_addr = ldsbase + (unsigned)(row * 260 + c4 * 4) * 4u;
            unsigned long long gaddr =
                (unsigned long long)(uintptr_t)(x + (size_t)grow * F_IN + c4 * 4);
            asm volatile("global_load_async_to_lds_b128 %0, %1, off"
                         :: "v"(lds_addr), "v"(gaddr) : "memory");
        }
        asm volatile("s_wait_asynccnt 0x0" ::: "memory");
    }
    __syncthreads();

    const int lane  = tid & 31;
    const int w     = tid >> 5;
    const int wm    = w >> 2;            // 0..1 : 16-row band
    const int wn    = w & 3;             // 0..3 : 64-col band
    const int l16   = lane & 15;
    const int khalf = (lane >> 4) << 1;  // 0 or 2 (A/B fragment K split across lane halves)

    const float* arow = &xs[wm * 16 + l16][0];

    // B pointers: each 16-col subtile lies fully inside one head (boundary every 32 cols)
    const float* bbase[4];
#pragma unroll
    for (int s = 0; s < 4; ++s) {
        int col = wn * 64 + s * 16 + l16;
        int h = col >> 5, d = col & 31;
        bbase[s] = W + (size_t)h * (F_IN * DHEAD) + d;
    }

    v8f cacc[4];
#pragma unroll
    for (int s = 0; s < 4; ++s) cacc[s] = (v8f){};

    for (int k0 = 0; k0 < F_IN; k0 += 4) {
        const int kk = k0 + khalf;
        v2f a;
        a.x = arow[kk];
        a.y = arow[kk + 1];
#pragma unroll
        for (int s = 0; s < 4; ++s) {
            v2f b;
            b.x = bbase[s][(size_t)kk * DHEAD];
            b.y = bbase[s][(size_t)(kk + 1) * DHEAD];
            cacc[s] = __builtin_amdgcn_wmma_f32_16x16x4_f32(
                false, a, false, b, (short)0, cacc[s], false, false);
        }
    }

    // C/D layout: VGPR r -> lanes 0-15: (m=r, n=lane), lanes 16-31: (m=r+8, n=lane-16)
    const int mstore = m0b + wm * 16 + (lane >> 4) * 8;
    float* Hp = H + (size_t)mstore * F_OUT;
    if (m0b + 32 <= nrows) {                // block-uniform fast path (always true for N%32==0)
#pragma unroll
        for (int s = 0; s < 4; ++s) {
            const int col = wn * 64 + s * 16 + l16;
#pragma unroll
            for (int r = 0; r < 8; ++r)
                Hp[(size_t)r * F_OUT + col] = cacc[s][r];
        }
    } else {
#pragma unroll
        for (int s = 0; s < 4; ++s) {
            const int col = wn * 64 + s * 16 + l16;
#pragma unroll
            for (int r = 0; r < 8; ++r)
                if (mstore + r < nrows)
                    Hp[(size_t)r * F_OUT + col] = cacc[s][r];
        }
    }
}

// ---------- K2: per-node, per-head attention scores ----------
// block = 256 = 8 heads x 32 lanes; one node per block
__global__ __launch_bounds__(256) void gat_scores(
    const float* __restrict__ H, const float* __restrict__ a,
    float* __restrict__ sdst, float* __restrict__ ssrc)
{
    const int node = blockIdx.x;
    const int h = threadIdx.x >> 5, lane = threadIdx.x & 31;
    float val = H[(size_t)node * F_OUT + h * DHEAD + lane];
    float sd = val * a[h * 64 + lane];
    float ss = val * a[h * 64 + 32 + lane];
    for (int off = 16; off > 0; off >>= 1) {
        sd += __shfl_down(sd, off, 32);
        ss += __shfl_down(ss, off, 32);
    }
    if (lane == 0) {
        sdst[(size_t)node * HEADS + h] = sd;
        ssrc[(size_t)node * HEADS + h] = ss;
    }
}

// ---------- K3: segment max of edge logits (ordered-uint atomicMax) ----------
__global__ void gat_edge_max(const int* __restrict__ adj,
                             const float* __restrict__ sdst,
                             const float* __restrict__ ssrc,
                             unsigned* __restrict__ mkey, int E)
{
    int t = blockIdx.x * blockDim.x + threadIdx.x;
    if (t >= E * HEADS) return;
    int e = t >> 3, h = t & 7;
    int d = adj[e], s = adj[E + e];
    float z = lrelu(sdst[(size_t)d * HEADS + h] + ssrc[(size_t)s * HEADS + h]);
    atomicMax(&mkey[(size_t)d * HEADS + h], f_ord(z));
}

// ---------- K4: segment sum of exp(logit - max) ----------
__global__ void gat_edge_sum(const int* __restrict__ adj,
                             const float* __restrict__ sdst,
                             const float* __restrict__ ssrc,
                             const unsigned* __restrict__ mkey,
                             float* __restrict__ denom, int E)
{
    int t = blockIdx.x * blockDim.x + threadIdx.x;
    if (t >= E * HEADS) return;
    int e = t >> 3, h = t & 7;
    int d = adj[e], s = adj[E + e];
    float z  = lrelu(sdst[(size_t)d * HEADS + h] + ssrc[(size_t)s * HEADS + h]);
    float ex = __expf(z - f_unord(mkey[(size_t)d * HEADS + h]));
    unsafeAtomicAdd(&denom[(size_t)d * HEADS + h], ex);
}

// ---------- K5: weighted aggregation out[dst] += alpha * H[src] ----------
// one edge per block; 256 threads = 8 heads x 32 dims
__global__ __launch_bounds__(256) void gat_aggregate(
    const int* __restrict__ adj,
    const float* __restrict__ sdst, const float* __restrict__ ssrc,
    const unsigned* __restrict__ mkey, const float* __restrict__ denom,
    const float* __restrict__ H, float* __restrict__ out, int E)
{
    const int e = blockIdx.x;
    const int h = threadIdx.x >> 5, lane = threadIdx.x & 31;
    int d = adj[e], s = adj[E + e];
    float z     = lrelu(sdst[(size_t)d * HEADS + h] + ssrc[(size_t)s * HEADS + h]);
    float m     = f_unord(mkey[(size_t)d * HEADS + h]);
    float alpha = __expf(z - m) / (denom[(size_t)d * HEADS + h] + GAT_EPS);
    float val   = H[(size_t)s * F_OUT + h * DHEAD + lane];
    unsafeAtomicAdd(&out[(size_t)d * F_OUT + h * DHEAD + lane], alpha * val);
}

extern "C" void kernel_launch(void* const* d_in, const int* in_sizes, int n_in,
                              void* d_out, int out_size, void* d_ws, size_t ws_size,
                              hipStream_t stream) {
    (void)n_in; (void)ws_size;
    const float* x   = (const float*)d_in[0];
    const int*   adj = (const int*)d_in[1];
    const float* W   = (const float*)d_in[2];
    const float* a   = (const float*)d_in[3];
    float* out = (float*)d_out;

    const int N = in_sizes[0] / F_IN;   // 100000
    const int E = in_sizes[1] / 2;      // 1600000

    // workspace layout
    char* ws = (char*)d_ws;
    float*    H     = (float*)ws;                 size_t off = (size_t)N * F_OUT * sizeof(float);
    float*    sdst  = (float*)(ws + off);         off += (size_t)N * HEADS * sizeof(float);
    float*    ssrc  = (float*)(ws + off);         off += (size_t)N * HEADS * sizeof(float);
    unsigned* mkey  = (unsigned*)(ws + off);      off += (size_t)N * HEADS * sizeof(unsigned);
    float*    denom = (float*)(ws + off);

    hipMemsetAsync(mkey,  0, (size_t)N * HEADS * sizeof(unsigned), stream); // key 0 == -inf sentinel
    hipMemsetAsync(denom, 0, (size_t)N * HEADS * sizeof(float), stream);
    hipMemsetAsync(out,   0, (size_t)out_size * sizeof(float), stream);

    gat_gemm_wmma<<<(N + 31) / 32, 256, 0, stream>>>(x, W, H, N);
    gat_scores<<<N, 256, 0, stream>>>(H, a, sdst, ssrc);

    const int tot = E * HEADS;
    gat_edge_max<<<(tot + 255) / 256, 256, 0, stream>>>(adj, sdst, ssrc, mkey, E);
    gat_edge_sum<<<(tot + 255) / 256, 256, 0, stream>>>(adj, sdst, ssrc, mkey, denom, E);
    gat_aggregate<<<E, 256, 0, stream>>>(adj, sdst, ssrc, mkey, denom, H, out, E);
}